// VQDiffusionVAE_73581379715476
// MI455X (gfx1250) — compile-verified
//
#include <hip/hip_runtime.h>

typedef float v2f __attribute__((ext_vector_type(2)));
typedef float v4f __attribute__((ext_vector_type(4)));
typedef float v8f __attribute__((ext_vector_type(8)));

namespace {
constexpr int kBn   = 32;              // batch
constexpr int kD    = 4;               // code dim
constexpr int kHW   = 32 * 32;         // 1024
constexpr int kN    = kBn * kHW;       // 32768 vectors
constexpr int kBDHW = kBn * kD * kHW;  // 131072 elements per tensor output
constexpr int kK    = 8192;            // codebook entries
constexpr int kTilesPerWave  = 2;                         // 32 z-rows per wave
constexpr int kWaves         = kN / (16 * kTilesPerWave); // 1024
constexpr int kBlock         = 256;                       // 8 wave32s
constexpr int kWavesPerBlock = kBlock / 32;
constexpr int kGrid          = kWaves / kWavesPerBlock;   // 128
constexpr float kBeta = 0.25f;
// d_ws layout (floats): [0, kWaves) loss partials, [kWaves, kWaves+kK) cnorm
constexpr int kCnormOfs = kWaves;
}  // namespace

// ---- Pass 0: cnorm[k] = ||codebook[k]||^2 (one coalesced float4 per thread).
__global__ __launch_bounds__(256) void vq_cnorm_kernel(
    const float* __restrict__ cb, float* __restrict__ cnorm) {
  const int k = blockIdx.x * 256 + threadIdx.x;
  const v4f c = *(const v4f*)(cb + (size_t)k * 4);
  cnorm[k] = c.x * c.x + c.y * c.y + c.z * c.z + c.w * c.w;
}

// ---- Pass 1: one wave owns 2 tiles of 16 encoder vectors.
// A = (-2*z) tile (16x4 f32), B = 16 codebook rows (4x16),
// C = cnorm broadcast per column, so V_WMMA_F32_16X16X4_F32 yields
// dist2 - ||z||^2 directly (argmin-equivalent). Running per-column argmin is
// kept per lane and reduced across the 16-lane halves at the end.
__global__ __launch_bounds__(kBlock) void vq_argmin_kernel(
    const float* __restrict__ ze, const float* __restrict__ cb,
    const float* __restrict__ cnorm, float* __restrict__ out,
    float* __restrict__ partials) {
  const int lane = threadIdx.x & 31;
  const int lq   = lane & 15;
  const int half = lane >> 4;
  const int wave = blockIdx.x * kWavesPerBlock + (threadIdx.x >> 5);
  const int nbase = wave * (16 * kTilesPerWave);

  // One-shot cold-cache priming: each wave prefetches 32 distinct cachelines
  // of the codebook and of cnorm (collectively covering both arrays).
  {
    const int line  = (wave * 32 + lane) & 1023;  // 1024 x 128B lines = 128KB
    __builtin_prefetch(cb + (size_t)line * 32, 0, 3);
    const int line2 = (wave * 32 + lane) & 255;   // 256 x 128B lines = 32KB
    __builtin_prefetch(cnorm + (size_t)line2 * 32, 0, 3);
  }

  // ---- Load A tiles: lane lq holds row (tile*16+lq); halves hold dims {0,1}/{2,3}.
  // z_e is [B,4,H,W]; flat row n -> (b = n>>10, r = n&1023), dim stride = 1024.
  v2f a[kTilesPerWave];
#pragma unroll
  for (int t = 0; t < kTilesPerWave; ++t) {
    const int n  = nbase + t * 16 + lq;
    const int bi = n >> 10;
    const int r  = n & (kHW - 1);
    const float* p = ze + (size_t)bi * (kD * kHW) + (2 * half) * kHW + r;
    a[t].x = -2.0f * p[0];
    a[t].y = -2.0f * p[kHW];
  }

  v8f best[kTilesPerWave], bestk[kTilesPerWave];
#pragma unroll
  for (int t = 0; t < kTilesPerWave; ++t)
#pragma unroll
    for (int i = 0; i < 8; ++i) { best[t][i] = 3.0e38f; bestk[t][i] = 0.0f; }

  float kf = (float)lq;  // column index this lane scores (exact in f32, k<8192)
#pragma unroll 2
  for (int kb = 0; kb < kK; kb += 16) {
    // B chunk: contiguous float2 per lane -> 256B coalesced per wave.
    const v2f bvec = *(const v2f*)(cb + (size_t)(kb + lq) * 4 + 2 * half);
    // Precomputed ||c||^2: 16 unique dwords per wave (halves duplicate).
    const float cn = cnorm[kb + lq];
    v8f cmat;
#pragma unroll
    for (int i = 0; i < 8; ++i) cmat[i] = cn;

#pragma unroll
    for (int t = 0; t < kTilesPerWave; ++t) {
      // D = (-2 z) * c + ||c||^2  == dist2 - ||z||^2
      v8f d = __builtin_amdgcn_wmma_f32_16x16x4_f32(
          false, a[t], false, bvec, (short)0, cmat, false, false);
#pragma unroll
      for (int i = 0; i < 8; ++i) {
        const bool take = d[i] < best[t][i];  // strict <: keep first (lowest k)
        best[t][i]  = take ? d[i] : best[t][i];
        bestk[t][i] = take ? kf   : bestk[t][i];
      }
    }
    kf += 16.0f;
  }

  // ---- Argmin across the 16 columns (lanes) of each half, tie-break lower k.
#pragma unroll
  for (int mask = 1; mask <= 8; mask <<= 1) {
#pragma unroll
    for (int t = 0; t < kTilesPerWave; ++t)
#pragma unroll
      for (int i = 0; i < 8; ++i) {
        const float ob = __shfl_xor(best[t][i], mask, 32);
        const float ok = __shfl_xor(bestk[t][i], mask, 32);
        const bool take = (ob < best[t][i]) || (ob == best[t][i] && ok < bestk[t][i]);
        best[t][i]  = take ? ob : best[t][i];
        bestk[t][i] = take ? ok : bestk[t][i];
      }
  }

  // ---- Gather + write. Lanes with lq<8 own row (lq + 8*half) of each tile.
  float s = 0.0f;
  if (lq < 8) {
#pragma unroll
    for (int t = 0; t < kTilesPerWave; ++t) {
      float bk = bestk[t][0];
#pragma unroll
      for (int i = 1; i < 8; ++i) bk = (lq == i) ? bestk[t][i] : bk;
      const int kidx = (int)bk;
      const int row  = lq + 8 * half;
      const int n    = nbase + t * 16 + row;
      const int bi   = n >> 10;
      const int r    = n & (kHW - 1);
      const v4f cq = *(const v4f*)(cb + (size_t)kidx * 4);  // 16B-aligned gather
#pragma unroll
      for (int dd = 0; dd < 4; ++dd) {
        const int off  = bi * (kD * kHW) + dd * kHW + r;
        const float zev  = ze[off];
        const float zq   = cq[dd];
        const float diff = zq - zev;           // explicit rounding for STE
        out[off]         = zev + diff;         // z_q_st
        out[kBDHW + off] = zq;                 // z_q
        s += diff * diff;
      }
      out[2 * kBDHW + n] = bk;                 // indices (concatenated as f32)
    }
  }
  // Deterministic per-wave loss partial (no float atomics).
#pragma unroll
  for (int mask = 1; mask <= 16; mask <<= 1) s += __shfl_xor(s, mask, 32);
  if (lane == 0) partials[wave] = s;
}

// ---- Pass 2: fixed-tree loss reduction (replay-deterministic).
__global__ __launch_bounds__(256) void vq_loss_kernel(
    const float* __restrict__ partials, float* __restrict__ out) {
  __shared__ float sm[256];
  float s = 0.0f;
  for (int i = threadIdx.x; i < kWaves; i += 256) s += partials[i];
  sm[threadIdx.x] = s;
  __syncthreads();
  for (int st = 128; st > 0; st >>= 1) {
    if ((int)threadIdx.x < st) sm[threadIdx.x] += sm[threadIdx.x + st];
    __syncthreads();
  }
  if (threadIdx.x == 0)
    out[2 * kBDHW + kN] = (1.0f + kBeta) * sm[0] / (float)kBDHW;  // le + beta*lcommit
}

extern "C" void kernel_launch(void* const* d_in, const int* in_sizes, int n_in,
                              void* d_out, int out_size, void* d_ws, size_t ws_size,
                              hipStream_t stream) {
  const float* ze = (const float*)d_in[0];   // z_e  [32,4,32,32] f32
  const float* cb = (const float*)d_in[1];   // codebook [8192,4] f32
  float* out      = (float*)d_out;           // [z_q_st | z_q | indices | loss]
  float* ws       = (float*)d_ws;
  float* partials = ws;                      // kWaves floats
  float* cnorm    = ws + kCnormOfs;          // kK floats

  vq_cnorm_kernel<<<kK / 256, 256, 0, stream>>>(cb, cnorm);
  vq_argmin_kernel<<<kGrid, kBlock, 0, stream>>>(ze, cb, cnorm, out, partials);
  vq_loss_kernel<<<1, 256, 0, stream>>>(partials, out);
}